// NeuronPool_46840913330745
// MI455X (gfx1250) — compile-verified
//
#include <hip/hip_runtime.h>
#include <math.h>
#include <stdint.h>

// Problem constants (from reference): B=4, S=4096, D=2048, N=8192, K=8
#define DDIM   2048
#define NNEUR  8192
#define MROWS  16384   // B*S
#define TOPK   8

typedef __attribute__((ext_vector_type(2))) float v2f;
typedef __attribute__((ext_vector_type(8))) float v8f;

// Async global->LDS copy of 16 bytes (CDNA5 GLOBAL_LOAD_ASYNC_TO_LDS_B128,
// tracked by ASYNCcnt; LDS address = wave-relative offset in a VGPR).
__device__ __forceinline__ void async_copy_b128(const void* gptr, void* lptr) {
    unsigned lds_off = (unsigned)(uintptr_t)lptr;   // addr[31:0] = LDS offset (ISA 10.2)
    asm volatile("global_load_async_to_lds_b128 %0, %1, off"
                 :: "v"(lds_off), "v"(gptr) : "memory");
}
__device__ __forceinline__ void wait_async(int n) {
    if (n == 0) asm volatile("s_wait_asynccnt 0x0" ::: "memory");
    else        asm volatile("s_wait_asynccnt 0x2" ::: "memory");
}

// ---------------------------------------------------------------------------
// Kernel A: q[M][D] = x[M][D] @ W_q[D][D]^T + b_q        (C = A * Bt^T + bias)
// BM=64, BN=64, BK=16; 256 threads = 8 waves in 2(M) x 4(N); each wave: two
// 16x16 fp32 WMMA C-frags. K-chunks double-buffered via async-to-LDS loads.
// ---------------------------------------------------------------------------
#define QBM 64
#define QBN 64
#define QBK 16
#define QLDPAD 20   // padded LDS row stride (floats)
#define NCHUNK (DDIM / QBK)

__global__ __launch_bounds__(256)
void qgemm_kernel(const float* __restrict__ A, const float* __restrict__ Bt,
                  const float* __restrict__ bias, float* __restrict__ C)
{
    __shared__ float sA[2][QBM][QLDPAD];
    __shared__ float sB[2][QBN][QLDPAD];

    const int tid  = threadIdx.x;
    const int lane = tid & 31;
    const int wave = tid >> 5;
    const int wm   = wave >> 2;          // 0..1  (32 rows each)
    const int wn   = wave & 3;           // 0..3  (16 cols each)
    const int m0   = blockIdx.y * QBM;
    const int n0   = blockIdx.x * QBN;

    const int lo = lane & 15;
    const int hi = lane >> 4;            // 0 or 1

    // staging assignment: one float4 of A and of Bt per thread per chunk
    const int sr = tid >> 2;             // row 0..63
    const int sk = (tid & 3) * 4;        // k sub-offset 0,4,8,12
    const float* gA = A  + (size_t)(m0 + sr) * DDIM + sk;
    const float* gB = Bt + (size_t)(n0 + sr) * DDIM + sk;

    v8f c0 = {}; v8f c1 = {};

    // prologue: async-stage chunk 0 into buffer 0
    async_copy_b128(gA, &sA[0][sr][sk]);
    async_copy_b128(gB, &sB[0][sr][sk]);

    for (int i = 0; i < NCHUNK; ++i) {
        const int buf = i & 1;
        if (i + 1 < NCHUNK) {            // prefetch next chunk into other buffer
            const int kn = (i + 1) * QBK;
            async_copy_b128(gA + kn, &sA[buf ^ 1][sr][sk]);
            async_copy_b128(gB + kn, &sB[buf ^ 1][sr][sk]);
            wait_async(2);               // oldest 2 (current chunk) have landed
        } else {
            wait_async(0);
        }
        __syncthreads();

        const int ma = wm * 32 + lo;
        const int nb = wn * 16 + lo;
#pragma unroll
        for (int kk = 0; kk < QBK; kk += 4) {
            const int kb = kk + 2 * hi;
            v2f a0 = { sA[buf][ma][kb],      sA[buf][ma][kb + 1] };
            v2f a1 = { sA[buf][ma + 16][kb], sA[buf][ma + 16][kb + 1] };
            v2f b  = { sB[buf][nb][kb],      sB[buf][nb][kb + 1] };
            c0 = __builtin_amdgcn_wmma_f32_16x16x4_f32(false, a0, false, b, (short)0, c0, false, false);
            c1 = __builtin_amdgcn_wmma_f32_16x16x4_f32(false, a1, false, b, (short)0, c1, false, false);
        }
        __syncthreads();
    }

    // store with fused bias; C frag: VGPR r -> M = r + 8*hi, N = lo
    const int colg  = n0 + wn * 16 + lo;
    const float bv  = bias[colg];
    const int rbase = m0 + wm * 32 + 8 * hi;
#pragma unroll
    for (int r = 0; r < 8; ++r) {
        C[(size_t)(rbase + r)      * DDIM + colg] = c0[r] + bv;
        C[(size_t)(rbase + 16 + r) * DDIM + colg] = c1[r] + bv;
    }
}

// ---------------------------------------------------------------------------
// Kernel B: fused scores + top-8 + softmax.
// One workgroup per 16 rows; q-tile async-staged into LDS. 8 waves each own
// groups of 4 N-tiles: 4 independent 16x16 fp32 WMMA accumulation chains per
// wave over K=2048 (shared A-frag). Frags dumped to LDS; lanes 0..15 keep a
// sorted per-row top-8; cross-wave merge + softmax -> idx/weights outputs.
// ---------------------------------------------------------------------------
#define QTLD 2052                       // 2048 + 4 pad (floats)
#define SC_LD 65                        // 16x64 frag staging row stride
#define NTIL 4                          // N-tiles per wave per pass

__global__ __launch_bounds__(256)
void scores_topk_kernel(const float* __restrict__ Q, const float* __restrict__ Nr,
                        int* __restrict__ idx_out, float* __restrict__ w_out)
{
    extern __shared__ float sm[];
    float* sq  = sm;                                    // 16 * QTLD
    float* ssc = sm + 16 * QTLD;                        // 8 * 16 * SC_LD
    float* mbs = ssc + 8 * 16 * SC_LD;                  // 8 * 16 * 8 scores
    int*   mbi = reinterpret_cast<int*>(mbs + 8 * 16 * TOPK);

    const int tid  = threadIdx.x;
    const int lane = tid & 31;
    const int wave = tid >> 5;
    const int lo   = lane & 15;
    const int hi   = lane >> 4;
    const int m0   = blockIdx.x * 16;

    // async-stage q tile: 16 x 2048 floats (32 b128 transfers per thread)
    for (int i = tid; i < 16 * (DDIM / 4); i += 256) {
        const int r  = i >> 9;            // / 512
        const int c4 = i & 511;
        async_copy_b128(Q + (size_t)(m0 + r) * DDIM + c4 * 4,
                        sq + r * QTLD + c4 * 4);
    }
    wait_async(0);
    __syncthreads();

    // per-row top-8 state (valid in lanes 0..15; row = m0 + lane)
    float ts[TOPK];
    int   ti[TOPK];
#pragma unroll
    for (int j = 0; j < TOPK; ++j) { ts[j] = -INFINITY; ti[j] = 0; }

    const float* arow = sq + lo * QTLD + 2 * hi;
    float* sw = ssc + wave * 16 * SC_LD;

    for (int tg = wave * NTIL; tg < NNEUR / 16; tg += 8 * NTIL) {  // 16 uniform iters
        const int nb = tg * 16;                                    // first of 64 columns
        const float* b0 = Nr + (size_t)(nb + lo) * DDIM + 2 * hi;
        const float* b1 = b0 + (size_t)16 * DDIM;
        const float* b2 = b0 + (size_t)32 * DDIM;
        const float* b3 = b0 + (size_t)48 * DDIM;
        __builtin_prefetch(Nr + (size_t)(nb + 64 * 8 + lo) * DDIM, 0, 1);

        v8f c0 = {}, c1 = {}, c2 = {}, c3 = {};
#pragma unroll 4
        for (int k = 0; k < DDIM; k += 4) {
            v2f a = { arow[k], arow[k + 1] };
            c0 = __builtin_amdgcn_wmma_f32_16x16x4_f32(false, a, false, (v2f){ b0[k], b0[k+1] }, (short)0, c0, false, false);
            c1 = __builtin_amdgcn_wmma_f32_16x16x4_f32(false, a, false, (v2f){ b1[k], b1[k+1] }, (short)0, c1, false, false);
            c2 = __builtin_amdgcn_wmma_f32_16x16x4_f32(false, a, false, (v2f){ b2[k], b2[k+1] }, (short)0, c2, false, false);
            c3 = __builtin_amdgcn_wmma_f32_16x16x4_f32(false, a, false, (v2f){ b3[k], b3[k+1] }, (short)0, c3, false, false);
        }

        // dump frags: row = r + 8*hi, tile t at columns 16t..16t+15
#pragma unroll
        for (int r = 0; r < 8; ++r) {
            float* rowp = sw + (r + 8 * hi) * SC_LD + lo;
            rowp[0]  = c0[r];
            rowp[16] = c1[r];
            rowp[32] = c2[r];
            rowp[48] = c3[r];
        }
        __syncthreads();

        if (lane < 16) {
            const float* rowp = sw + lane * SC_LD;
#pragma unroll
            for (int cc = 0; cc < 64; ++cc) {
                const float v = rowp[cc];
                if (v > ts[TOPK - 1]) {                // strict > : lowest-index tie-break
                    int j = TOPK - 1;
                    while (j > 0 && v > ts[j - 1]) {
                        ts[j] = ts[j - 1]; ti[j] = ti[j - 1]; --j;
                    }
                    ts[j] = v; ti[j] = nb + cc;
                }
            }
        }
        __syncthreads();
    }

    // publish per-wave top-8, then lanes of wave 0 merge + softmax
    if (lane < 16) {
        const int base = (wave * 16 + lane) * TOPK;
#pragma unroll
        for (int j = 0; j < TOPK; ++j) { mbs[base + j] = ts[j]; mbi[base + j] = ti[j]; }
    }
    __syncthreads();

    if (tid < 16) {
        float fs[TOPK]; int fi[TOPK];
#pragma unroll
        for (int j = 0; j < TOPK; ++j) { fs[j] = -INFINITY; fi[j] = 0; }
        for (int w = 0; w < 8; ++w) {
            const int base = (w * 16 + tid) * TOPK;
#pragma unroll
            for (int j = 0; j < TOPK; ++j) {
                const float v = mbs[base + j];
                if (v > fs[TOPK - 1]) {
                    int p = TOPK - 1;
                    while (p > 0 && v > fs[p - 1]) { fs[p] = fs[p - 1]; fi[p] = fi[p - 1]; --p; }
                    fs[p] = v; fi[p] = mbi[base + j];
                }
            }
        }
        // softmax over descending fs (fs[0] is max)
        float e[TOPK]; float sum = 0.f;
#pragma unroll
        for (int j = 0; j < TOPK; ++j) { e[j] = __expf(fs[j] - fs[0]); sum += e[j]; }
        const float inv = 1.f / sum;
        const size_t ob = (size_t)(m0 + tid) * TOPK;
#pragma unroll
        for (int j = 0; j < TOPK; ++j) { idx_out[ob + j] = fi[j]; w_out[ob + j] = e[j] * inv; }
    }
}

// ---------------------------------------------------------------------------
// Kernel C: output[m,:] = sum_j w[m,j] * neurons[idx[m,j], :]
// One wave per row; float4 gather (neurons are L2-resident).
// ---------------------------------------------------------------------------
__global__ __launch_bounds__(256)
void gather_out_kernel(const float* __restrict__ Nr, const int* __restrict__ idxv,
                       const float* __restrict__ wv, float* __restrict__ out)
{
    const int lane = threadIdx.x & 31;
    const int wave = threadIdx.x >> 5;
    const int m    = blockIdx.x * 8 + wave;

    float w[TOPK];
    const float* rows[TOPK];
#pragma unroll
    for (int j = 0; j < TOPK; ++j) {
        w[j]    = wv[(size_t)m * TOPK + j];
        rows[j] = Nr + (size_t)idxv[(size_t)m * TOPK + j] * DDIM;
    }
    for (int c = lane * 4; c < DDIM; c += 32 * 4) {
        float4 acc = make_float4(0.f, 0.f, 0.f, 0.f);
#pragma unroll
        for (int j = 0; j < TOPK; ++j) {
            float4 v = *reinterpret_cast<const float4*>(rows[j] + c);
            acc.x += w[j] * v.x; acc.y += w[j] * v.y;
            acc.z += w[j] * v.z; acc.w += w[j] * v.w;
        }
        *reinterpret_cast<float4*>(out + (size_t)m * DDIM + c) = acc;
    }
}

// ---------------------------------------------------------------------------
extern "C" void kernel_launch(void* const* d_in, const int* in_sizes, int n_in,
                              void* d_out, int out_size, void* d_ws, size_t ws_size,
                              hipStream_t stream)
{
    const float* x       = (const float*)d_in[0];   // [4,4096,2048]
    const float* neurons = (const float*)d_in[1];   // [8192,2048]
    const float* W_q     = (const float*)d_in[2];   // [2048,2048]
    const float* b_q     = (const float*)d_in[3];   // [2048]
    // d_in[4] = k scalar (8), compile-time constant here

    float* out_o   = (float*)d_out;                                   // [M, D]
    int*   out_idx = (int*)((float*)d_out + (size_t)MROWS * DDIM);    // [M, K] int32 bits
    float* out_w   = (float*)d_out + (size_t)MROWS * DDIM
                                   + (size_t)MROWS * TOPK;            // [M, K]

    float* q_ws = (float*)d_ws;   // 16384*2048*4 = 128 MiB scratch for q

    // Kernel A: q = x @ W_q^T + b_q
    dim3 gA(DDIM / QBN, MROWS / QBM);   // (32, 256)
    qgemm_kernel<<<gA, 256, 0, stream>>>(x, W_q, b_q, q_ws);

    // Kernel B: fused scores + top-8 + softmax
    const size_t smemB = (size_t)(16 * QTLD + 8 * 16 * SC_LD + 8 * 16 * TOPK) * 4
                       + (size_t)(8 * 16 * TOPK) * 4;
    scores_topk_kernel<<<MROWS / 16, 256, smemB, stream>>>(q_ws, neurons, out_idx, out_w);

    // Kernel C: weighted gather
    gather_out_kernel<<<MROWS / 8, 256, 0, stream>>>(neurons, out_idx, out_w, out_o);
}